// SpatialNonIntersectionAxiom_46480136077416
// MI455X (gfx1250) — compile-verified
//
#include <hip/hip_runtime.h>
#include <hip/hip_bf16.h>

#define E_N 4096
#define EPSILON_F 0.001f
#define PROX_F 0.15f

typedef float v2f __attribute__((ext_vector_type(2)));
typedef float v4f __attribute__((ext_vector_type(4)));
typedef float v8f __attribute__((ext_vector_type(8)));
typedef int   v4i __attribute__((ext_vector_type(4)));

// Workspace layout (32-bit word offsets into d_ws)
enum {
  OF_PSX = 0,
  OF_PSY  = OF_PSX  + E_N,
  OF_D1X  = OF_PSY  + E_N,
  OF_D1Y  = OF_D1X  + E_N,
  OF_DPS  = OF_D1Y  + E_N,   // d1 . ps
  OF_A    = OF_DPS  + E_N,   // max(|d1|^2, 1e-12)
  OF_INVA = OF_A    + E_N,
  OF_HL   = OF_INVA + E_N,   // 0.5*sqrt(|d1|^2)
  OF_MIDX = OF_HL   + E_N,
  OF_MIDY = OF_MIDX + E_N,
  OF_MID2 = OF_MIDY + E_N,
  OF_SRC  = OF_MID2 + E_N,   // int
  OF_DST  = OF_SRC  + E_N,   // int
  OF_SUM  = OF_DST  + E_N,   // float accumulator
  OF_CNT  = OF_SUM  + 1      // uint accumulator
};

__global__ void nx_init_acc(float* __restrict__ w) {
  w[OF_SUM] = 0.0f;
  ((unsigned*)w)[OF_CNT] = 0u;
}

__global__ void nx_prep(const float* __restrict__ pos,
                        const int* __restrict__ eidx,
                        float* __restrict__ w) {
  int e = blockIdx.x * blockDim.x + threadIdx.x;
  if (e >= E_N) return;
  int s = eidx[e];
  int d = eidx[E_N + e];
  float sx = pos[2 * s], sy = pos[2 * s + 1];
  float tx = pos[2 * d], ty = pos[2 * d + 1];
  float ex = tx - sx, ey = ty - sy;               // d1
  float len2 = ex * ex + ey * ey;
  float a = fmaxf(len2, 1e-12f);
  float mx = sx + 0.5f * ex, my = sy + 0.5f * ey; // midpoint
  w[OF_PSX + e] = sx;  w[OF_PSY + e] = sy;
  w[OF_D1X + e] = ex;  w[OF_D1Y + e] = ey;
  w[OF_DPS + e] = ex * sx + ey * sy;
  w[OF_A + e] = a;
  w[OF_INVA + e] = __builtin_amdgcn_rcpf(a);
  w[OF_HL + e] = 0.5f * __builtin_amdgcn_sqrtf(len2);
  w[OF_MIDX + e] = mx; w[OF_MIDY + e] = my;
  w[OF_MID2 + e] = mx * mx + my * my;
  ((int*)w)[OF_SRC + e] = s;
  ((int*)w)[OF_DST + e] = d;
}

// Select element v (compile-time constant under full unroll) from a lo/hi v4f pair.
#define ELT_F(LO, HI, V) ((V) < 4 ? (LO)[(V)] : (HI)[(V) - 4])
#define ELT_I(LO, HI, V) ((V) < 4 ? (LO)[(V)] : (HI)[(V) - 4])

// One wave32 per 16x16 tile of edge pairs. Four V_WMMA_F32_16X16X4_F32 per
// tile produce b, c, f, md^2; epilogue does the clamp/rcp/sqrt chain.
__global__ __launch_bounds__(32) void nx_pair_tiles(float* __restrict__ w) {
  const int ti = blockIdx.y, tj = blockIdx.x;
  if (tj < ti) return;                  // uniform per-block: EXEC stays all-1
  const int lane = threadIdx.x;
  const int n = lane & 15;
  const bool lo = (lane < 16);
  const int i0 = ti << 4, j0 = tj << 4;
  const int ia = i0 + n, ja = j0 + n;

  const float* psx = w + OF_PSX;  const float* psy = w + OF_PSY;
  const float* d1x = w + OF_D1X;  const float* d1y = w + OF_D1Y;
  const float* dps = w + OF_DPS;  const float* aA  = w + OF_A;
  const float* inva = w + OF_INVA; const float* hl = w + OF_HL;
  const float* mdx = w + OF_MIDX; const float* mdy = w + OF_MIDY;
  const float* md2 = w + OF_MID2;
  const int* srcA = (const int*)w + OF_SRC;
  const int* dstA = (const int*)w + OF_DST;

  // Row-side (A) scalars for row ia, col-side (B) scalars for col ja.
  float d1x_i = d1x[ia], d1y_i = d1y[ia], dps_i = dps[ia];
  float psx_i = psx[ia], psy_i = psy[ia];
  float mx_i = mdx[ia], my_i = mdy[ia], m2_i = md2[ia];
  float d1x_j = d1x[ja], d1y_j = d1y[ja], dps_j = dps[ja];
  float psx_j = psx[ja], psy_j = psy[ja];
  float mx_j = mdx[ja], my_j = mdy[ja], m2_j = md2[ja];

  // A 16x4: lanes 0-15 -> K=0,1 ; lanes 16-31 -> K=2,3. B 4x16 mirrored.
  v2f bA = { lo ? d1x_i : 0.0f,          lo ? d1y_i : 0.0f };
  v2f bB = { lo ? d1x_j : 0.0f,          lo ? d1y_j : 0.0f };
  v2f cA = { lo ? d1x_i : dps_i,         lo ? d1y_i : 0.0f };
  v2f cB = { lo ? -psx_j : 1.0f,         lo ? -psy_j : 0.0f };
  v2f fA = { lo ? psx_i : 1.0f,          lo ? psy_i : 0.0f };
  v2f fB = { lo ? d1x_j : -dps_j,        lo ? d1y_j : 0.0f };
  v2f mA = { lo ? -2.0f * mx_i : m2_i,   lo ? -2.0f * my_i : 1.0f };
  v2f mB = { lo ? mx_j : 1.0f,           lo ? my_j : m2_j };

  v8f z = {0.0f, 0.0f, 0.0f, 0.0f, 0.0f, 0.0f, 0.0f, 0.0f};
  v8f Bt = __builtin_amdgcn_wmma_f32_16x16x4_f32(false, bA, false, bB, (short)0, z, false, false);
  v8f Ct = __builtin_amdgcn_wmma_f32_16x16x4_f32(false, cA, false, cB, (short)0, z, false, false);
  v8f Ft = __builtin_amdgcn_wmma_f32_16x16x4_f32(false, fA, false, fB, (short)0, z, false, false);
  v8f Mt = __builtin_amdgcn_wmma_f32_16x16x4_f32(false, mA, false, mB, (short)0, z, false, false);

  // Epilogue: this lane owns col j = ja, rows rb..rb+7 (contiguous, 16B aligned).
  const int j = ja;
  const float a_j = aA[j], inva_j = inva[j], hl_j = hl[j];
  const int src_j = srcA[j], dst_j = dstA[j];

  const int rb = i0 + (lo ? 0 : 8);
  const v4f aLo   = *(const v4f*)(aA   + rb), aHi   = *(const v4f*)(aA   + rb + 4);
  const v4f iaLo  = *(const v4f*)(inva + rb), iaHi  = *(const v4f*)(inva + rb + 4);
  const v4f hlLo  = *(const v4f*)(hl   + rb), hlHi  = *(const v4f*)(hl   + rb + 4);
  const v4f pxLo  = *(const v4f*)(psx  + rb), pxHi  = *(const v4f*)(psx  + rb + 4);
  const v4f pyLo  = *(const v4f*)(psy  + rb), pyHi  = *(const v4f*)(psy  + rb + 4);
  const v4f dxLo  = *(const v4f*)(d1x  + rb), dxHi  = *(const v4f*)(d1x  + rb + 4);
  const v4f dyLo  = *(const v4f*)(d1y  + rb), dyHi  = *(const v4f*)(d1y  + rb + 4);
  const v4i srLo  = *(const v4i*)(srcA + rb), srHi  = *(const v4i*)(srcA + rb + 4);
  const v4i dsLo  = *(const v4i*)(dstA + rb), dsHi  = *(const v4i*)(dstA + rb + 4);

  float lsum = 0.0f;
  unsigned lcnt = 0u;
#pragma unroll
  for (int v = 0; v < 8; ++v) {
    const int i = rb + v;
    const float b = Bt[v], c = Ct[v], f = Ft[v], q = Mt[v]; // q = md^2
    const float a_i    = ELT_F(aLo, aHi, v);
    const float inva_i = ELT_F(iaLo, iaHi, v);
    const float hl_i   = ELT_F(hlLo, hlHi, v);
    const int   src_i  = ELT_I(srLo, srHi, v);
    const int   dst_i  = ELT_I(dsLo, dsHi, v);
    const bool adj = (src_i == src_j) || (src_i == dst_j) ||
                     (dst_i == src_j) || (dst_i == dst_j);
    const float thr = hl_i + hl_j + PROX_F;
    const bool msk = (j > i) && (!adj) && (q < thr * thr);

    const float denom = fmaxf(a_i * a_j - b * b, 1e-12f);
    float s = (b * f - c * a_j) * __builtin_amdgcn_rcpf(denom);
    s = fminf(fmaxf(s, 0.0f), 1.0f);
    float t = (b * s + f) * inva_j;
    t = fminf(fmaxf(t, 0.0f), 1.0f);
    s = (b * t - c) * inva_i;
    s = fminf(fmaxf(s, 0.0f), 1.0f);

    const float ddx = (ELT_F(pxLo, pxHi, v) - psx_j) + s * ELT_F(dxLo, dxHi, v) - t * d1x_j;
    const float ddy = (ELT_F(pyLo, pyHi, v) - psy_j) + s * ELT_F(dyLo, dyHi, v) - t * d1y_j;
    const float dist = __builtin_amdgcn_sqrtf(ddx * ddx + ddy * ddy);

    lsum += msk ? fmaxf(EPSILON_F - dist, 0.0f) : 0.0f;
    lcnt += msk ? 1u : 0u;
  }

  // wave32 tree reduction
  for (int off = 16; off > 0; off >>= 1) {
    lsum += __shfl_down(lsum, off, 32);
    lcnt += __shfl_down(lcnt, off, 32);
  }
  if (lane == 0) {
    atomicAdd(w + OF_SUM, lsum);
    atomicAdd((unsigned*)w + OF_CNT, lcnt);
  }
}

__global__ void nx_finalize(const float* __restrict__ w, float* __restrict__ out) {
  unsigned c = ((const unsigned*)w)[OF_CNT];
  if (c < 1u) c = 1u;
  out[0] = w[OF_SUM] / (float)c;
}

extern "C" void kernel_launch(void* const* d_in, const int* in_sizes, int n_in,
                              void* d_out, int out_size, void* d_ws, size_t ws_size,
                              hipStream_t stream) {
  (void)in_sizes; (void)n_in; (void)out_size; (void)ws_size;
  const float* pos = (const float*)d_in[0];   // (1, 4096, 2) f32
  // d_in[1] adjacency: unused by the reference — never read (saves 67 MB of HBM traffic)
  const int* eidx = (const int*)d_in[2];      // (2, 4096) int
  float* w = (float*)d_ws;
  float* out = (float*)d_out;

  nx_init_acc<<<1, 1, 0, stream>>>(w);
  nx_prep<<<E_N / 256, 256, 0, stream>>>(pos, eidx, w);
  dim3 grid(E_N / 16, E_N / 16);               // tj = x, ti = y; lower triangle exits
  nx_pair_tiles<<<grid, 32, 0, stream>>>(w);
  nx_finalize<<<1, 1, 0, stream>>>(w, out);
}